// GGNNLayer_38878043963480
// MI455X (gfx1250) — compile-verified
//
#include <hip/hip_runtime.h>

#define N_NODES 100000
#define N_EDGES 1280000
#define HID 64
#define MSG 64
#define N_ETYPE 8
#define G3 192

typedef __attribute__((ext_vector_type(2))) float v2f;
typedef __attribute__((ext_vector_type(8))) float v8f;

__device__ __forceinline__ v8f wmma_f32(v2f a, v2f b, v8f c) {
    // V_WMMA_F32_16X16X4_F32 : D = A(16x4) * B(4x16) + C(16x16), all fp32
    return __builtin_amdgcn_wmma_f32_16x16x4_f32(
        /*neg_a=*/false, a, /*neg_b=*/false, b,
        /*c_mod=*/(short)0, c, /*reuse_a=*/false, /*reuse_b=*/false);
}

// ---------------------------------------------------------------------------
// Zero the m_agg accumulator (harness poisons ws; must re-zero every call).
// ---------------------------------------------------------------------------
__global__ void zero_kernel(float* __restrict__ p, int n) {
    int idx = (blockIdx.x * blockDim.x + threadIdx.x) * 4;
    if (idx < n) {
        float4 z = make_float4(0.f, 0.f, 0.f, 0.f);
        *(float4*)(p + idx) = z;
    }
}

// ---------------------------------------------------------------------------
// Ht[t][n][m] = sum_h A[t][m][h] * h[n][h]
// One wave per (16-node tile, edge type). fp32 WMMA, K=64 in 16 steps of 4.
// ---------------------------------------------------------------------------
__global__ void __launch_bounds__(32) ht_kernel(const float* __restrict__ h,
                                                const float* __restrict__ em,
                                                float* __restrict__ Ht) {
    const int lane = threadIdx.x;
    const int n0   = blockIdx.x * 16;
    const int t    = blockIdx.y;
    const int row  = lane & 15;          // A: M index / B: N index
    const int kb   = (lane >> 4) * 2;    // K pair selected by lane half
    const int rbase = (lane >> 4) * 8;   // D: M offset for lane half

    // Preload A fragments: 16 nodes x 64 hid (each lane: 32 floats)
    v2f a[16];
    const float* hrow = h + (size_t)(n0 + row) * HID;
#pragma unroll
    for (int k = 0; k < 16; ++k)
        a[k] = *(const v2f*)(hrow + k * 4 + kb);

    const float* At = em + (size_t)t * (MSG * HID);
    float* outbase = Ht + ((size_t)t * N_NODES + n0) * MSG;

#pragma unroll
    for (int m0i = 0; m0i < 4; ++m0i) {
        const int m0 = m0i * 16;
        const float* brow = At + (size_t)(m0 + row) * HID;  // B col = A[t] row m0+row
        v8f acc = {};
#pragma unroll
        for (int k = 0; k < 16; ++k) {
            v2f b = *(const v2f*)(brow + k * 4 + kb);
            acc = wmma_f32(a[k], b, acc);
        }
#pragma unroll
        for (int r = 0; r < 8; ++r)
            outbase[(size_t)(r + rbase) * MSG + m0 + row] = acc[r];
    }
}

// ---------------------------------------------------------------------------
// Per-edge gather from Ht + float atomic scatter-add into m_agg.
// 16 threads per edge -> one coalesced 256B row read, 4-wide fp32 atomics.
// ---------------------------------------------------------------------------
__global__ void edge_kernel(const float* __restrict__ Ht,
                            const int* __restrict__ esrc,
                            const int* __restrict__ edst,
                            const int* __restrict__ etyp,
                            float* __restrict__ magg) {
    int gid = blockIdx.x * blockDim.x + threadIdx.x;
    int e   = gid >> 4;
    int sub = (gid & 15) * 4;
    if (e >= N_EDGES) return;
    int t = etyp[e];
    int s = esrc[e];
    int d = edst[e];
    const float4 v = *(const float4*)(Ht + ((size_t)t * N_NODES + s) * MSG + sub);
    float* dst = magg + (size_t)d * MSG + sub;
    atomicAdd(dst + 0, v.x);
    atomicAdd(dst + 1, v.y);
    atomicAdd(dst + 2, v.z);
    atomicAdd(dst + 3, v.w);
}

// ---------------------------------------------------------------------------
// GRU update: gi = m_agg @ w_ih^T + b_ih ; gh = h @ w_hh^T + b_hh ; gate.
// One wave per 16-node tile; 12 gi + 12 gh WMMA tiles staged through LDS.
// ---------------------------------------------------------------------------
__global__ void __launch_bounds__(32) gru_kernel(const float* __restrict__ h,
                                                 const float* __restrict__ magg,
                                                 const float* __restrict__ wih,
                                                 const float* __restrict__ whh,
                                                 const float* __restrict__ bih,
                                                 const float* __restrict__ bhh,
                                                 float* __restrict__ out) {
    __shared__ float lgi[16 * G3];
    __shared__ float lgh[16 * G3];
    const int lane  = threadIdx.x;
    const int n0    = blockIdx.x * 16;
    const int row   = lane & 15;
    const int kb    = (lane >> 4) * 2;
    const int rbase = (lane >> 4) * 8;

    // Preload A fragments for both GEMMs
    v2f aM[16], aH[16];
    const float* mrow = magg + (size_t)(n0 + row) * HID;
    const float* hrow = h   + (size_t)(n0 + row) * HID;
#pragma unroll
    for (int k = 0; k < 16; ++k) {
        aM[k] = *(const v2f*)(mrow + k * 4 + kb);
        aH[k] = *(const v2f*)(hrow + k * 4 + kb);
    }

#pragma unroll
    for (int j = 0; j < 12; ++j) {
        const float* bi = wih + (size_t)(j * 16 + row) * HID;
        const float* bh = whh + (size_t)(j * 16 + row) * HID;
        v8f ai = {}, ah = {};
#pragma unroll
        for (int k = 0; k < 16; ++k) {
            v2f b0 = *(const v2f*)(bi + k * 4 + kb);
            ai = wmma_f32(aM[k], b0, ai);
            v2f b1 = *(const v2f*)(bh + k * 4 + kb);
            ah = wmma_f32(aH[k], b1, ah);
        }
#pragma unroll
        for (int r = 0; r < 8; ++r) {
            lgi[(r + rbase) * G3 + j * 16 + row] = ai[r];
            lgh[(r + rbase) * G3 + j * 16 + row] = ah[r];
        }
    }
    __syncthreads();

    // Gating: 16 nodes x 64 cols = 1024 elems, 32 per lane, coalesced stores.
#pragma unroll 4
    for (int i = 0; i < 32; ++i) {
        int idx = i * 32 + lane;
        int nl  = idx >> 6;
        int c   = idx & 63;
        float gir = lgi[nl * G3 + c]        + bih[c];
        float giz = lgi[nl * G3 + 64 + c]   + bih[64 + c];
        float gin = lgi[nl * G3 + 128 + c]  + bih[128 + c];
        float ghr = lgh[nl * G3 + c]        + bhh[c];
        float ghz = lgh[nl * G3 + 64 + c]   + bhh[64 + c];
        float ghn = lgh[nl * G3 + 128 + c]  + bhh[128 + c];
        float r = 1.f / (1.f + __expf(-(gir + ghr)));
        float z = 1.f / (1.f + __expf(-(giz + ghz)));
        float n = tanhf(gin + r * ghn);
        float hv = h[(size_t)(n0 + nl) * HID + c];
        out[(size_t)(n0 + nl) * HID + c] = (1.f - z) * n + z * hv;
    }
}

extern "C" void kernel_launch(void* const* d_in, const int* in_sizes, int n_in,
                              void* d_out, int out_size, void* d_ws, size_t ws_size,
                              hipStream_t stream) {
    const float* h   = (const float*)d_in[0];
    const float* em  = (const float*)d_in[1];
    const float* wih = (const float*)d_in[2];
    const float* whh = (const float*)d_in[3];
    const float* bih = (const float*)d_in[4];
    const float* bhh = (const float*)d_in[5];
    const int* esrc  = (const int*)d_in[6];
    const int* edst  = (const int*)d_in[7];
    const int* etyp  = (const int*)d_in[8];
    float* out = (float*)d_out;

    float* Ht   = (float*)d_ws;                            // 8*100000*64 f32 = 204.8 MB
    float* magg = Ht + (size_t)N_ETYPE * N_NODES * MSG;    // 100000*64 f32  =  25.6 MB

    const int nZero = N_NODES * HID;  // 6.4M floats, exact multiple of 4*256
    zero_kernel<<<(nZero / 4 + 255) / 256, 256, 0, stream>>>(magg, nZero);

    dim3 g1(N_NODES / 16, N_ETYPE);   // 6250 x 8 waves
    ht_kernel<<<g1, 32, 0, stream>>>(h, em, Ht);

    edge_kernel<<<(N_EDGES * 16) / 256, 256, 0, stream>>>(Ht, esrc, edst, etyp, magg);

    gru_kernel<<<N_NODES / 16, 32, 0, stream>>>(h, magg, wih, whh, bih, bhh, out);
}